// InvariantInteraction_5738076308126
// MI455X (gfx1250) — compile-verified
//
#include <hip/hip_runtime.h>
#include <math.h>

typedef __attribute__((ext_vector_type(2))) float v2f;
typedef __attribute__((ext_vector_type(8))) float v8f;

#define CDIM 128
#define NLM  16
#define KEL  4

// ---------------------------------------------------------------------------
// Kernel 1: x = node_feats @ W_up * (1/sqrt(C))     [N,128] x [128,128]
// One wave per 16x16 output tile; 8 waves/block cover the full 128-wide row.
// Two independent WMMA accumulation chains to hide the f32 16x16x4 latency.
// ---------------------------------------------------------------------------
__global__ void k_linear_up(const float* __restrict__ A,   // node_feats [N,128]
                            const float* __restrict__ W,   // W_up [128,128]
                            float* __restrict__ X,         // out x [N,128]
                            float scale)
{
    const int wave = threadIdx.x >> 5;     // 0..7 -> column tile
    const int lane = threadIdx.x & 31;
    const int m0 = blockIdx.x * 16;
    const int n0 = wave * 16;
    const int m  = lane & 15;              // row within tile (A) / col (B,D)
    const int kh = lane >> 4;              // K-half selector

    const float* __restrict__ arow = A + (size_t)(m0 + m) * CDIM;

    v8f acc0 = {}; v8f acc1 = {};
    for (int k0 = 0; k0 < CDIM; k0 += 8) {
        const int ka = k0 + kh * 2;
        v2f a0 = { arow[ka],     arow[ka + 1] };
        v2f a1 = { arow[ka + 4], arow[ka + 5] };
        const int bo = ka * CDIM + n0 + m;
        v2f b0 = { W[bo],            W[bo + CDIM] };
        v2f b1 = { W[bo + 4*CDIM],   W[bo + 5*CDIM] };
        acc0 = __builtin_amdgcn_wmma_f32_16x16x4_f32(false, a0, false, b0,
                                                     (short)0, acc0, false, false);
        acc1 = __builtin_amdgcn_wmma_f32_16x16x4_f32(false, a1, false, b1,
                                                     (short)0, acc1, false, false);
    }
    const int nn = n0 + m;
    #pragma unroll
    for (int r = 0; r < 8; ++r)
        X[(size_t)(m0 + r + 8*kh) * CDIM + nn] = (acc0[r] + acc1[r]) * scale;
}

// ---------------------------------------------------------------------------
// Kernel 2: W_comb[e][l] = W_lin[l] @ W_skip[e][l] * combScale
// 16 small 128x128x128 GEMMs; grid = (e,l,row-tile), waves = col-tiles.
// ---------------------------------------------------------------------------
__global__ void k_combine(const float* __restrict__ Wl,   // [4,128,128]
                          const float* __restrict__ Ws,   // [4,4,128,128]
                          float* __restrict__ Wc,         // [4,4,128,128]
                          float scale)
{
    const int b  = blockIdx.x;             // 0..127
    const int mt = b & 7;
    const int l  = (b >> 3) & 3;
    const int e  = b >> 5;
    const int wave = threadIdx.x >> 5;
    const int lane = threadIdx.x & 31;
    const int m0 = mt * 16, n0 = wave * 16;
    const int m  = lane & 15, kh = lane >> 4;

    const float* __restrict__ Am = Wl + (size_t)l * CDIM * CDIM + (size_t)(m0 + m) * CDIM;
    const float* __restrict__ Bm = Ws + (size_t)(e * 4 + l) * CDIM * CDIM;

    v8f acc0 = {}; v8f acc1 = {};
    for (int k0 = 0; k0 < CDIM; k0 += 8) {
        const int ka = k0 + kh * 2;
        v2f a0 = { Am[ka],     Am[ka + 1] };
        v2f a1 = { Am[ka + 4], Am[ka + 5] };
        const int bo = ka * CDIM + n0 + m;
        v2f b0 = { Bm[bo],          Bm[bo + CDIM] };
        v2f b1 = { Bm[bo + 4*CDIM], Bm[bo + 5*CDIM] };
        acc0 = __builtin_amdgcn_wmma_f32_16x16x4_f32(false, a0, false, b0,
                                                     (short)0, acc0, false, false);
        acc1 = __builtin_amdgcn_wmma_f32_16x16x4_f32(false, a1, false, b1,
                                                     (short)0, acc1, false, false);
    }
    float* __restrict__ Dm = Wc + (size_t)(e * 4 + l) * CDIM * CDIM;
    #pragma unroll
    for (int r = 0; r < 8; ++r)
        Dm[(size_t)(m0 + r + 8*kh) * CDIM + n0 + m] = (acc0[r] + acc1[r]) * scale;
}

// ---------------------------------------------------------------------------
// Kernel 3: edge tensor product + scatter-add (memory/atomics bound).
// One block (256 threads) per edge: thread = (lm-half, channel).
// msg[recv, lm, c] += x[send, c] * edge_attrs[e, lm] * edge_feats[e, l(lm), c]
// ---------------------------------------------------------------------------
__global__ void k_edge_scatter(const float* __restrict__ x,      // [N,128]
                               const float* __restrict__ eattr,  // [E,16]
                               const float* __restrict__ efeat,  // [E,4,128]
                               const int*   __restrict__ eidx,   // [2,E]
                               float* __restrict__ msg,          // [N,16,128]
                               int E)
{
    const int e = blockIdx.x;
    const int t = threadIdx.x;          // 0..255
    const int c    = t & 127;
    const int half = t >> 7;            // lm 0..7 or 8..15

    __shared__ float sattr[NLM];
    if (t < NLM) sattr[t] = eattr[(size_t)e * NLM + t];
    __syncthreads();

    const int recv = eidx[e];
    const int send = eidx[E + e];

    const float xs = x[(size_t)send * CDIM + c];
    const float* __restrict__ R = efeat + (size_t)e * KEL * CDIM + c;
    const float R0 = R[0 * CDIM];
    const float R1 = R[1 * CDIM];
    const float R2 = R[2 * CDIM];
    const float R3 = R[3 * CDIM];

    float* __restrict__ dst = msg + ((size_t)recv * NLM) * CDIM + c;
    #pragma unroll
    for (int j = 0; j < 8; ++j) {
        const int lm = half * 8 + j;
        const int l  = (lm == 0) ? 0 : (lm < 4) ? 1 : (lm < 9) ? 2 : 3;
        const float Rv = (l == 0) ? R0 : (l == 1) ? R1 : (l == 2) ? R2 : R3;
        atomicAdd(dst + (size_t)lm * CDIM, xs * sattr[lm] * Rv);
    }
}

// ---------------------------------------------------------------------------
// Kernel 4: out[n,lm,:] = sum_e attr[n,e] * ( msg[n,lm,:] @ W_comb[e][l(lm)] )
// One block per (node-tile, lm); 8 waves = column tiles. Four independent
// WMMA accumulator chains (one per chemical element) -> good ILP.
// ---------------------------------------------------------------------------
__global__ void k_out(const float* __restrict__ msg,    // [N,16,128]
                      const float* __restrict__ attrs,  // [N,4]
                      const float* __restrict__ Wc,     // [4,4,128,128]
                      float* __restrict__ out)          // [N,16,128]
{
    const int lm = blockIdx.x & 15;
    const int mt = blockIdx.x >> 4;
    const int l  = (lm == 0) ? 0 : (lm < 4) ? 1 : (lm < 9) ? 2 : 3;
    const int wave = threadIdx.x >> 5;
    const int lane = threadIdx.x & 31;
    const int m0 = mt * 16, n0 = wave * 16;
    const int m  = lane & 15, kh = lane >> 4;

    const float* __restrict__ arow = msg + ((size_t)(m0 + m) * NLM + lm) * CDIM;
    const float w0 = attrs[(size_t)(m0 + m) * KEL + 0];
    const float w1 = attrs[(size_t)(m0 + m) * KEL + 1];
    const float w2 = attrs[(size_t)(m0 + m) * KEL + 2];
    const float w3 = attrs[(size_t)(m0 + m) * KEL + 3];

    const float* __restrict__ B0 = Wc + (size_t)(0 * 4 + l) * CDIM * CDIM;
    const float* __restrict__ B1 = Wc + (size_t)(1 * 4 + l) * CDIM * CDIM;
    const float* __restrict__ B2 = Wc + (size_t)(2 * 4 + l) * CDIM * CDIM;
    const float* __restrict__ B3 = Wc + (size_t)(3 * 4 + l) * CDIM * CDIM;

    v8f acc0 = {}, acc1 = {}, acc2 = {}, acc3 = {};
    for (int k0 = 0; k0 < CDIM; k0 += 4) {
        const int ka = k0 + kh * 2;
        const v2f a = { arow[ka], arow[ka + 1] };
        const int bo = ka * CDIM + n0 + m;

        v2f ae, b;
        ae = a * w0; b = (v2f){ B0[bo], B0[bo + CDIM] };
        acc0 = __builtin_amdgcn_wmma_f32_16x16x4_f32(false, ae, false, b,
                                                     (short)0, acc0, false, false);
        ae = a * w1; b = (v2f){ B1[bo], B1[bo + CDIM] };
        acc1 = __builtin_amdgcn_wmma_f32_16x16x4_f32(false, ae, false, b,
                                                     (short)0, acc1, false, false);
        ae = a * w2; b = (v2f){ B2[bo], B2[bo + CDIM] };
        acc2 = __builtin_amdgcn_wmma_f32_16x16x4_f32(false, ae, false, b,
                                                     (short)0, acc2, false, false);
        ae = a * w3; b = (v2f){ B3[bo], B3[bo + CDIM] };
        acc3 = __builtin_amdgcn_wmma_f32_16x16x4_f32(false, ae, false, b,
                                                     (short)0, acc3, false, false);
    }
    const int nn = n0 + m;
    #pragma unroll
    for (int r = 0; r < 8; ++r)
        out[((size_t)(m0 + r + 8*kh) * NLM + lm) * CDIM + nn] =
            acc0[r] + acc1[r] + acc2[r] + acc3[r];
}

// ---------------------------------------------------------------------------
extern "C" void kernel_launch(void* const* d_in, const int* in_sizes, int n_in,
                              void* d_out, int out_size, void* d_ws, size_t ws_size,
                              hipStream_t stream)
{
    (void)n_in; (void)out_size; (void)ws_size;

    const float* node_attrs = (const float*)d_in[0];   // [N,4]
    const float* node_feats = (const float*)d_in[1];   // [N,128]
    const float* edge_attrs = (const float*)d_in[2];   // [E,16]
    const float* edge_feats = (const float*)d_in[3];   // [E,4,128]
    const int*   edge_index = (const int*)  d_in[4];   // [2,E]
    const float* W_up       = (const float*)d_in[5];   // [128,128]
    const float* W_lin      = (const float*)d_in[6];   // [4,128,128]
    const float* W_skip     = (const float*)d_in[7];   // [4,4,128,128]
    float* out = (float*)d_out;                        // [N,16,128]

    const int N = in_sizes[1] / CDIM;
    const int E = in_sizes[2] / NLM;

    // workspace layout
    float* x    = (float*)d_ws;                         // [N,128]
    float* msg  = x   + (size_t)N * CDIM;               // [N,16,128]
    float* Wc   = msg + (size_t)N * NLM * CDIM;         // [4,4,128,128]

    const float inv_sqrt_c  = 1.0f / sqrtf((float)CDIM);
    const float inv_sqrt_kc = 1.0f / sqrtf((float)(KEL * CDIM));
    const float comb_scale  = inv_sqrt_c * inv_sqrt_kc / 16.0f; // /AVG_NUM_NEIGHBORS

    hipMemsetAsync(msg, 0, (size_t)N * NLM * CDIM * sizeof(float), stream);

    k_linear_up<<<N / 16, 256, 0, stream>>>(node_feats, W_up, x, inv_sqrt_c);
    k_combine  <<<128,    256, 0, stream>>>(W_lin, W_skip, Wc, comb_scale);
    k_edge_scatter<<<E,   256, 0, stream>>>(x, edge_attrs, edge_feats,
                                            edge_index, msg, E);
    k_out<<<(N / 16) * NLM, 256, 0, stream>>>(msg, node_attrs, Wc, out);
}